// VoxelGraph_67997922230587
// MI455X (gfx1250) — compile-verified
//
#include <hip/hip_runtime.h>
#include <stdint.h>

// ---------------------------------------------------------------------------
// VoxelGraph neighbor search for MI455X (gfx1250, wave32).
// Grid constants derived from the reference (PC_RANGE / VOXEL_SIZE / N_BATCH):
//   dims = [*, 251, 251, 26]; batch multiplier uses only dims[1..3].
// ---------------------------------------------------------------------------
static constexpr int DXv = 251, DYv = 251, DZv = 26;
static constexpr int NBATCH = 4;
static constexpr int NB_PER_B = DXv * DYv * DZv;          // 1,638,026
static constexpr int NBINS = NBATCH * NB_PER_B;           // 6,552,104
static constexpr int SCAN_T = 256;
static constexpr int SCAN_I = 16;
static constexpr int SCAN_TILE = SCAN_T * SCAN_I;         // 4096
static constexpr int KSLOT = 32;

__device__ __forceinline__ int clampi(int v, int lo, int hi) {
  return v < lo ? lo : (v > hi ? hi : v);
}

// Matches jnp.floor((p - pc_min)/vsize) in f32 (true IEEE division, no recip).
__device__ __forceinline__ int voxel_id(float b, float x, float y, float z) {
  int vb = (int)floorf(b / 0.999f);
  int vx = (int)floorf((x - (-50.0f)) / 0.4f);
  int vy = (int)floorf((y - (-50.0f)) / 0.4f);
  int vz = (int)floorf((z - (-5.0f)) / 0.4f);
  vb = clampi(vb, 0, NBATCH - 1);                 // defensive (data is in-range)
  vx = clampi(vx, 0, DXv - 1);
  vy = clampi(vy, 0, DYv - 1);
  vz = clampi(vz, 0, DZv - 1);
  return ((vb * DXv + vx) * DYv + vy) * DZv + vz;
}

// ---- 1) histogram -----------------------------------------------------------
__global__ void vg_hist(const float4* __restrict__ ref, uint32_t* __restrict__ binptr, int n) {
  int i = blockIdx.x * blockDim.x + threadIdx.x;
  if (i >= n) return;
  float4 p = ref[i];                               // (b, x, y, z)
  atomicAdd(&binptr[voxel_id(p.x, p.y, p.z, p.w)], 1u);
}

// ---- 2) exclusive scan over NBINS (in place), 3 kernels ---------------------
__global__ void vg_scan_tile(uint32_t* __restrict__ data, uint32_t* __restrict__ bsums, int n) {
  __shared__ uint32_t lds[SCAN_TILE];
  __shared__ uint32_t tsum[SCAN_T];
  const int tid = threadIdx.x;
  const int base = blockIdx.x * SCAN_TILE;
#pragma unroll
  for (int i = 0; i < SCAN_I; ++i) {
    int idx = base + i * SCAN_T + tid;
    lds[i * SCAN_T + tid] = (idx < n) ? data[idx] : 0u;
  }
  __syncthreads();
  uint32_t s = 0;
#pragma unroll
  for (int i = 0; i < SCAN_I; ++i) s += lds[tid * SCAN_I + i];
  tsum[tid] = s;
  __syncthreads();
  for (int off = 1; off < SCAN_T; off <<= 1) {      // Hillis-Steele inclusive
    uint32_t t = (tid >= off) ? tsum[tid - off] : 0u;
    __syncthreads();
    tsum[tid] += t;
    __syncthreads();
  }
  uint32_t run = (tid > 0) ? tsum[tid - 1] : 0u;    // exclusive thread offset
  uint32_t blocktot = tsum[SCAN_T - 1];
#pragma unroll
  for (int i = 0; i < SCAN_I; ++i) {                // serial exclusive within chunk
    uint32_t v = lds[tid * SCAN_I + i];
    lds[tid * SCAN_I + i] = run;
    run += v;
  }
  __syncthreads();
#pragma unroll
  for (int i = 0; i < SCAN_I; ++i) {
    int idx = base + i * SCAN_T + tid;
    if (idx < n) data[idx] = lds[i * SCAN_T + tid];
  }
  if (tid == 0) bsums[blockIdx.x] = blocktot;
}

__global__ void vg_scan_sums(uint32_t* __restrict__ sums, int n) {
  __shared__ uint32_t lds[SCAN_T];
  const int tid = threadIdx.x;
  uint32_t carry = 0;
  for (int base = 0; base < n; base += SCAN_T) {
    int idx = base + tid;
    uint32_t v = (idx < n) ? sums[idx] : 0u;
    lds[tid] = v;
    __syncthreads();
    for (int off = 1; off < SCAN_T; off <<= 1) {
      uint32_t t = (tid >= off) ? lds[tid - off] : 0u;
      __syncthreads();
      lds[tid] += t;
      __syncthreads();
    }
    uint32_t excl = (tid > 0) ? lds[tid - 1] : 0u;
    uint32_t tot = lds[SCAN_T - 1];
    __syncthreads();
    if (idx < n) sums[idx] = carry + excl;
    carry += tot;
  }
}

__global__ void vg_scan_add(uint32_t* __restrict__ data, const uint32_t* __restrict__ bsums, int n) {
  int idx = blockIdx.x * blockDim.x + threadIdx.x;
  if (idx < n) data[idx] += bsums[idx >> 12];       // 4096-element tiles
}

// ---- 3) scatter: binptr[v] morphs from start[v] into end[v] -----------------
__global__ void vg_scatter(const float4* __restrict__ ref, uint32_t* __restrict__ binptr,
                           uint32_t* __restrict__ sorted, int n) {
  int i = blockIdx.x * blockDim.x + threadIdx.x;
  if (i >= n) return;
  float4 p = ref[i];
  uint32_t pos = atomicAdd(&binptr[voxel_id(p.x, p.y, p.z, p.w)], 1u);
  sorted[pos] = (uint32_t)i;
}

// ---- 4) restore stable (ascending index) order inside each bin --------------
__global__ void vg_fixsort(const uint32_t* __restrict__ binptr, uint32_t* __restrict__ sorted, int nbins) {
  int v = blockIdx.x * blockDim.x + threadIdx.x;
  if (v >= nbins) return;
  uint32_t s = (v > 0) ? binptr[v - 1] : 0u;        // after scatter: binptr[v] == end[v]
  uint32_t e = binptr[v];
  if (e - s < 2u) return;
  for (uint32_t i = s + 1; i < e; ++i) {            // bins are tiny (Poisson mean ~0.08)
    uint32_t val = sorted[i];
    uint32_t j = i;
    while (j > s && sorted[j - 1] > val) { sorted[j] = sorted[j - 1]; --j; }
    sorted[j] = val;
  }
}

// ---- 5) one wave32 per query: 27 buckets on lanes, 32 slots on lanes --------
__global__ void vg_query(const float4* __restrict__ qry, const float4* __restrict__ ref,
                         const uint32_t* __restrict__ binptr, const uint32_t* __restrict__ sorted,
                         float* __restrict__ out, int nq, int nref) {
  const int lane = threadIdx.x & 31;
  const int qi = blockIdx.x * (blockDim.x >> 5) + (threadIdx.x >> 5);
  if (qi >= nq) return;

  float4 p = qry[qi];
  const int vb = clampi((int)floorf(p.x / 0.999f), 0, NBATCH - 1);
  const int vx = (int)floorf((p.y - (-50.0f)) / 0.4f);
  const int vy = (int)floorf((p.z - (-50.0f)) / 0.4f);
  const int vz = (int)floorf((p.w - (-5.0f)) / 0.4f);

  // lanes 0..26 own the 27 offsets in reference meshgrid(ij) order: dz fastest
  int startv = 0, cnt = 0;
  if (lane < 27) {
    int nx = vx + lane / 9 - 1;
    int ny = vy + (lane / 3) % 3 - 1;
    int nz = vz + lane % 3 - 1;
    if (nx >= 0 && nx < DXv && ny >= 0 && ny < DYv && nz >= 0 && nz < DZv) {
      int vid = ((vb * DXv + nx) * DYv + ny) * DZv + nz;
      int e = (int)binptr[vid];
      int s = (vid > 0) ? (int)binptr[vid - 1] : 0;
      startv = s;
      cnt = e - s;
    }
  }

  // wave inclusive scan (== reference cumsum; lanes >=27 contribute 0,
  // so csum[27..31] == total, which also pads the searchsorted below)
  int csum = cnt;
#pragma unroll
  for (int d = 1; d < 32; d <<= 1) {
    int t = __shfl_up(csum, d);
    if (lane >= d) csum += t;
  }
  const int total = __shfl(csum, 31);

  const int k = lane;                                // output slot
  const bool valid = k < total;

  // bucket = searchsorted(csum, k, 'right'): branchless binary search via shuffles
  int b = 0;
#pragma unroll
  for (int s = 16; s > 0; s >>= 1) {
    int t = b + s;
    int c = __shfl(csum, t - 1);
    if (c <= k) b = t;
  }
  const int bcl = b < 26 ? b : 26;
  int pm = __shfl(csum, b > 0 ? b - 1 : 0);
  const int prev = (b > 0) ? pm : 0;
  const int sb = __shfl(startv, bcl);
  int gidx = sb + (k - prev);
  gidx = clampi(gidx, 0, nref - 1);

  int r = -1;
  float sq = 0.0f;
  if (valid) {
    r = (int)sorted[gidx];
    float4 rp = ref[r];
    float dx = p.y - rp.y, dy = p.z - rp.z, dz = p.w - rp.w;
    sq = dx * dx + dy * dy + dz * dz;
  }

  // coalesced 128B stores per wave: out layout [e_ref | e_query | sqdist | valid]
  const size_t NQK = (size_t)nq * KSLOT;
  const size_t o = (size_t)qi * KSLOT + lane;
  out[o]            = (float)r;
  out[NQK + o]      = valid ? (float)qi : -1.0f;
  out[2 * NQK + o]  = sq;
  out[3 * NQK + o]  = valid ? 1.0f : 0.0f;
}

// ---------------------------------------------------------------------------
extern "C" void kernel_launch(void* const* d_in, const int* in_sizes, int n_in,
                              void* d_out, int out_size, void* d_ws, size_t ws_size,
                              hipStream_t stream) {
  (void)n_in; (void)out_size; (void)ws_size;
  const float4* ref = (const float4*)d_in[0];       // (N_REF, 4) float32
  const float4* qry = (const float4*)d_in[1];       // (N_QUERY, 4) float32
  const int nref = in_sizes[0] / 4;
  const int nq = in_sizes[1] / 4;

  // Workspace layout (~28.2 MB total; fits MI455X L2 entirely):
  uint32_t* binptr = (uint32_t*)d_ws;               // NBINS   (26.2 MB)
  uint32_t* sorted = binptr + NBINS;                // nref    ( 2.0 MB)
  const int nScan = (NBINS + SCAN_TILE - 1) / SCAN_TILE;  // 1600
  uint32_t* bsums = sorted + nref;                  // nScan   ( 6.4 KB)
  float* out = (float*)d_out;

  hipMemsetAsync(binptr, 0, (size_t)NBINS * sizeof(uint32_t), stream);
  vg_hist<<<(nref + 255) / 256, 256, 0, stream>>>(ref, binptr, nref);
  vg_scan_tile<<<nScan, SCAN_T, 0, stream>>>(binptr, bsums, NBINS);
  vg_scan_sums<<<1, SCAN_T, 0, stream>>>(bsums, nScan);
  vg_scan_add<<<(NBINS + 255) / 256, 256, 0, stream>>>(binptr, bsums, NBINS);
  vg_scatter<<<(nref + 255) / 256, 256, 0, stream>>>(ref, binptr, sorted, nref);
  vg_fixsort<<<(NBINS + 255) / 256, 256, 0, stream>>>(binptr, sorted, NBINS);
  vg_query<<<(nq + 7) / 8, 256, 0, stream>>>(qry, ref, binptr, sorted, out, nq, nref);
}